// Node_OP_21114059227218
// MI455X (gfx1250) — compile-verified
//
#include <hip/hip_runtime.h>
#include <math.h>

#define H 128
#define AST 132   // padded LDS row stride (floats): 132 mod 64 = 4 -> conflict-free A gathers, 16B aligned

typedef float v2f __attribute__((ext_vector_type(2)));
typedef float v8f __attribute__((ext_vector_type(8)));

__device__ __forceinline__ float atomic_add_f32(float* p, float v) {
    return __hip_atomic_fetch_add(p, v, __ATOMIC_RELAXED, __HIP_MEMORY_SCOPE_AGENT);
}

// K1: out = sigmoid(w0)*x0 + sigmoid(w1)*x1 ; zero agg ; zero BN accumulators
__global__ void gate_zero_kernel(const float* __restrict__ x0, const float* __restrict__ x1,
                                 const float* __restrict__ mw,
                                 float* __restrict__ out, float* __restrict__ agg,
                                 float* __restrict__ colsum, float* __restrict__ colsumsq,
                                 int n4) {
    int idx = blockIdx.x * blockDim.x + threadIdx.x;
    if (idx < n4) {
        float s0 = 1.0f / (1.0f + __expf(-mw[0]));
        float s1 = 1.0f / (1.0f + __expf(-mw[1]));
        float4 a = ((const float4*)x0)[idx];
        float4 b = ((const float4*)x1)[idx];
        float4 r;
        r.x = s0 * a.x + s1 * b.x;
        r.y = s0 * a.y + s1 * b.y;
        r.z = s0 * a.z + s1 * b.z;
        r.w = s0 * a.w + s1 * b.w;
        ((float4*)out)[idx] = r;
        ((float4*)agg)[idx] = make_float4(0.f, 0.f, 0.f, 0.f);
    }
    if (idx < H) { colsum[idx] = 0.f; colsumsq[idx] = 0.f; }
}

// K2: scatter-add: one wave per edge, each lane covers 4 columns
__global__ void scatter_kernel(const float* __restrict__ out, const int* __restrict__ ei,
                               float* __restrict__ agg, int E) {
    int gtid  = blockIdx.x * blockDim.x + threadIdx.x;
    int wave  = gtid >> 5;
    int lane  = gtid & 31;
    int nwave = (gridDim.x * blockDim.x) >> 5;
    for (int e = wave; e < E; e += nwave) {
        int src = ei[e];
        int dst = ei[E + e];
        float4 v = *(const float4*)(out + (size_t)src * H + lane * 4);
        float* p = agg + (size_t)dst * H + lane * 4;
        atomic_add_f32(p + 0, v.x);
        atomic_add_f32(p + 1, v.y);
        atomic_add_f32(p + 2, v.z);
        atomic_add_f32(p + 3, v.w);
    }
}

// K3: h = relu((out+agg)@W1 + b1)@W2 + b2 via V_WMMA_F32_16X16X4_F32
//     4 waves/block, 16 rows per wave, weights staged in LDS (W1 then W2).
//     Two n-tiles in flight per iteration: two independent WMMA chains hide
//     XDL latency while keeping VGPR usage < 256 (no vgpr-msb region).
__global__ __launch_bounds__(128)
void mlp_wmma_kernel(const float* __restrict__ out, const float* __restrict__ agg,
                     const float* __restrict__ W1, const float* __restrict__ b1,
                     const float* __restrict__ W2, const float* __restrict__ b2,
                     float* __restrict__ hout, int N) {
    __shared__ __align__(16) float sW[H * H];        // 64 KB, W1 then W2
    __shared__ __align__(16) float sA[4][16 * AST];  // per-wave input tile
    __shared__ __align__(16) float sT[4][16 * AST];  // per-wave relu(t) tile

    const int tid   = threadIdx.x;
    const int wave  = tid >> 5;
    const int lane  = tid & 31;
    const int l16   = lane & 15;
    const int khalf = (lane >> 4) << 1;          // 0 for lanes 0-15, 2 for lanes 16-31
    const int mrow  = (lane < 16) ? 0 : 8;       // C/D layout row base
    const int rowBase = blockIdx.x * 64 + wave * 16;

    // stage A = out + agg (16 rows x 128 cols per wave), zero-pad OOB rows
    for (int r = 0; r < 16; ++r) {
        int row = rowBase + r;
        float4 v = make_float4(0.f, 0.f, 0.f, 0.f);
        if (row < N) {
            float4 o = *(const float4*)(out + (size_t)row * H + lane * 4);
            float4 g = *(const float4*)(agg + (size_t)row * H + lane * 4);
            v = make_float4(o.x + g.x, o.y + g.y, o.z + g.z, o.w + g.w);
        }
        *(float4*)(&sA[wave][r * AST + lane * 4]) = v;
    }
    for (int i = tid * 4; i < H * H; i += 128 * 4)
        *(float4*)(&sW[i]) = *(const float4*)(W1 + i);
    __syncthreads();

    // GEMM1: t = relu(A @ W1 + b1) -> sT   (2 n-tiles per iteration)
    #pragma unroll 1
    for (int nt = 0; nt < 8; nt += 2) {
        const int nc0 = nt * 16 + l16;
        const int nc1 = nc0 + 16;
        const float bias0 = b1[nc0];
        const float bias1 = b1[nc1];
        v8f acc0, acc1;
        #pragma unroll
        for (int i = 0; i < 8; ++i) { acc0[i] = bias0; acc1[i] = bias1; }
        #pragma unroll 4
        for (int k0 = 0; k0 < H; k0 += 4) {
            const int kk = k0 + khalf;
            v2f a = *(const v2f*)(&sA[wave][l16 * AST + kk]);   // A: M=l16, K=kk,kk+1
            v2f bA, bB;
            bA.x = sW[kk * H + nc0];                            // B: K=kk,   N=nc0
            bA.y = sW[(kk + 1) * H + nc0];
            bB.x = sW[kk * H + nc1];
            bB.y = sW[(kk + 1) * H + nc1];
            acc0 = __builtin_amdgcn_wmma_f32_16x16x4_f32(false, a, false, bA,
                                                         (short)0, acc0, false, false);
            acc1 = __builtin_amdgcn_wmma_f32_16x16x4_f32(false, a, false, bB,
                                                         (short)0, acc1, false, false);
        }
        #pragma unroll
        for (int r = 0; r < 8; ++r) {
            float t0 = acc0[r] > 0.f ? acc0[r] : 0.f;
            float t1 = acc1[r] > 0.f ? acc1[r] : 0.f;
            sT[wave][(mrow + r) * AST + nc0] = t0;
            sT[wave][(mrow + r) * AST + nc1] = t1;
        }
    }
    __syncthreads();
    for (int i = tid * 4; i < H * H; i += 128 * 4)
        *(float4*)(&sW[i]) = *(const float4*)(W2 + i);
    __syncthreads();

    // GEMM2: h = T @ W2 + b2 -> global   (2 n-tiles per iteration)
    #pragma unroll 1
    for (int nt = 0; nt < 8; nt += 2) {
        const int nc0 = nt * 16 + l16;
        const int nc1 = nc0 + 16;
        const float bias0 = b2[nc0];
        const float bias1 = b2[nc1];
        v8f acc0, acc1;
        #pragma unroll
        for (int i = 0; i < 8; ++i) { acc0[i] = bias0; acc1[i] = bias1; }
        #pragma unroll 4
        for (int k0 = 0; k0 < H; k0 += 4) {
            const int kk = k0 + khalf;
            v2f a = *(const v2f*)(&sT[wave][l16 * AST + kk]);
            v2f bA, bB;
            bA.x = sW[kk * H + nc0];
            bA.y = sW[(kk + 1) * H + nc0];
            bB.x = sW[kk * H + nc1];
            bB.y = sW[(kk + 1) * H + nc1];
            acc0 = __builtin_amdgcn_wmma_f32_16x16x4_f32(false, a, false, bA,
                                                         (short)0, acc0, false, false);
            acc1 = __builtin_amdgcn_wmma_f32_16x16x4_f32(false, a, false, bB,
                                                         (short)0, acc1, false, false);
        }
        #pragma unroll
        for (int r = 0; r < 8; ++r) {
            int row = rowBase + mrow + r;
            if (row < N) {
                hout[(size_t)row * H + nc0] = acc0[r];
                hout[(size_t)row * H + nc1] = acc1[r];
            }
        }
    }
}

// K4: per-column sum / sumsq partial reduction
__global__ void colreduce_kernel(const float* __restrict__ h,
                                 float* __restrict__ colsum, float* __restrict__ colsumsq,
                                 int N) {
    int c = threadIdx.x;  // 128 threads = 128 columns
    float s = 0.f, q = 0.f;
    for (int row = blockIdx.x; row < N; row += gridDim.x) {
        float v = h[(size_t)row * H + c];
        s += v;
        q += v * v;
    }
    atomic_add_f32(&colsum[c], s);
    atomic_add_f32(&colsumsq[c], q);
}

// K5: fold mean/var into per-column scale/shift (biased variance, matches jnp.var)
__global__ void stats_kernel(const float* __restrict__ colsum, const float* __restrict__ colsumsq,
                             const float* __restrict__ gamma, const float* __restrict__ beta,
                             float* __restrict__ scl, float* __restrict__ shf, int N) {
    int c = threadIdx.x;
    float invN = 1.0f / (float)N;
    float mean = colsum[c] * invN;
    float var  = colsumsq[c] * invN - mean * mean;
    float inv  = rsqrtf(var + 1e-5f);
    float g    = gamma[c] * inv;
    scl[c] = g;
    shf[c] = beta[c] - mean * g;
}

// K6: in-place normalize
__global__ void bn_apply_kernel(float* __restrict__ h,
                                const float* __restrict__ scl, const float* __restrict__ shf,
                                int n4) {
    int idx = blockIdx.x * blockDim.x + threadIdx.x;
    if (idx >= n4) return;
    int c = (idx & 31) * 4;  // 32 float4 per 128-col row
    float4 v = ((const float4*)h)[idx];
    float4 s = *(const float4*)(scl + c);
    float4 t = *(const float4*)(shf + c);
    v.x = v.x * s.x + t.x;
    v.y = v.y * s.y + t.y;
    v.z = v.z * s.z + t.z;
    v.w = v.w * s.w + t.w;
    ((float4*)h)[idx] = v;
}

extern "C" void kernel_launch(void* const* d_in, const int* in_sizes, int n_in,
                              void* d_out, int out_size, void* d_ws, size_t ws_size,
                              hipStream_t stream) {
    const float* x0    = (const float*)d_in[0];
    const float* x1    = (const float*)d_in[1];
    const int*   ei    = (const int*)d_in[2];
    const float* mw    = (const float*)d_in[3];
    const float* W1    = (const float*)d_in[4];
    const float* b1    = (const float*)d_in[5];
    const float* W2    = (const float*)d_in[6];
    const float* b2    = (const float*)d_in[7];
    const float* gamma = (const float*)d_in[8];
    const float* beta  = (const float*)d_in[9];
    float* out = (float*)d_out;

    const int N = in_sizes[0] / H;     // 100000
    const int E = in_sizes[2] / 2;     // 1600000
    const int n4 = N * (H / 4);        // float4 element count

    float* agg      = (float*)d_ws;                 // N*H floats
    float* colsum   = agg + (size_t)N * H;          // 128
    float* colsumsq = colsum + H;                   // 128
    float* scl      = colsumsq + H;                 // 128
    float* shf      = scl + H;                      // 128

    gate_zero_kernel<<<(n4 + 255) / 256, 256, 0, stream>>>(x0, x1, mw, out, agg,
                                                           colsum, colsumsq, n4);
    scatter_kernel<<<4096, 256, 0, stream>>>(out, ei, agg, E);
    mlp_wmma_kernel<<<(N + 63) / 64, 128, 0, stream>>>(out, agg, W1, b1, W2, b2, out, N);
    colreduce_kernel<<<512, 128, 0, stream>>>(out, colsum, colsumsq, N);
    stats_kernel<<<1, 128, 0, stream>>>(colsum, colsumsq, gamma, beta, scl, shf, N);
    bn_apply_kernel<<<(n4 + 255) / 256, 256, 0, stream>>>(out, scl, shf, n4);
}